// OrthoLoss_20332375179379
// MI455X (gfx1250) — compile-verified
//
#include <hip/hip_runtime.h>
#include <hip/hip_bf16.h>

// ---------------------------------------------------------------------------
// OrthoLoss on MI455X (gfx1250): memory-bound fused softmax-KL reduction.
// 268 MB streamed @ 23.3 TB/s => ~11.5 us floor. No matmul => WMMA is not the
// right tool; instead use the CDNA5 async global->LDS pipeline
// (global_load_async_to_lds_b128 + s_wait_asynccnt) to double-buffer tiles.
// s_wait_dscnt 0x0 before each re-prefetch closes the LDS WAR race (async
// LDS writes are unordered vs the wave's own ds_loads per ISA 08 sec 4.1).
// ---------------------------------------------------------------------------

#define B_ROWS 65536
#define D_DIM 128
#define WAVES_PER_BLOCK 8
#define ROWS_PER_WAVE 4
#define NB1 (B_ROWS / (WAVES_PER_BLOCK * ROWS_PER_WAVE)) // 2048 blocks
#define NB2 (B_ROWS / 256)                               // 256 blocks

// ---------------------------------------------------------------------------
// Kernel 1: per-row KL terms for t and s branches.
// One wave (32 lanes) per row; 4 fp32 elements per lane (D=128).
// Chunks alternate t/s phase; each chunk stages 4 arrays x 512B into a
// per-wave LDS double buffer with async loads.
// ---------------------------------------------------------------------------
__global__ __launch_bounds__(256) void kl_rows_kernel(
    const float* __restrict__ mean_t, const float* __restrict__ mean_s,
    const float* __restrict__ log_std_t, const float* __restrict__ log_std_s,
    const float* __restrict__ noise_pt, const float* __restrict__ noise_ps,
    const float* __restrict__ noise_et, const float* __restrict__ noise_es,
    float* __restrict__ zt_part, float* __restrict__ zs_part)
{
    // [wave][buf][array][elem] : 8*2*4*128*4B = 32 KB
    __shared__ __align__(16) float lbuf[WAVES_PER_BLOCK][2][4][D_DIM];
    __shared__ float wz[WAVES_PER_BLOCK][2];

    const int tid  = threadIdx.x;
    const int w    = tid >> 5;
    const int lane = tid & 31;
    const long rowBase = ((long)blockIdx.x * WAVES_PER_BLOCK + w) * ROWS_PER_WAVE;

    const float* m_arr[2]  = {mean_t,    mean_s};
    const float* l_arr[2]  = {log_std_t, log_std_s};
    const float* ne_arr[2] = {noise_et,  noise_es};
    const float* np_arr[2] = {noise_pt,  noise_ps};

    // Low 32 bits of the generic address of a __shared__ object are the
    // group-segment byte offset (what the async instruction's VDST expects).
    const unsigned slotW = (unsigned)(size_t)&lbuf[w][0][0][0];

    float zAcc0 = 0.0f, zAcc1 = 0.0f;
    const int NCH = ROWS_PER_WAVE * 2;

#define ASYNC_PREFETCH(cc) do {                                                   \
        const int p_ = (cc) & 1;                                                  \
        const size_t eoff_ = ((size_t)(rowBase + ((cc) >> 1))) * D_DIM            \
                             + (size_t)lane * 4;                                  \
        const unsigned la_ = slotW + (unsigned)(((cc) & 1) * 2048)                \
                             + (unsigned)(lane * 16);                             \
        asm volatile("global_load_async_to_lds_b128 %0, %1, off"                  \
                     :: "v"(la_),          "v"(m_arr[p_]  + eoff_) : "memory");   \
        asm volatile("global_load_async_to_lds_b128 %0, %1, off"                  \
                     :: "v"(la_ + 512u),   "v"(l_arr[p_]  + eoff_) : "memory");   \
        asm volatile("global_load_async_to_lds_b128 %0, %1, off"                  \
                     :: "v"(la_ + 1024u),  "v"(ne_arr[p_] + eoff_) : "memory");   \
        asm volatile("global_load_async_to_lds_b128 %0, %1, off"                  \
                     :: "v"(la_ + 1536u),  "v"(np_arr[p_] + eoff_) : "memory");   \
    } while (0)

    ASYNC_PREFETCH(0);
    for (int c = 0; c < NCH; ++c) {
        if (c + 1 < NCH) {
            // WAR fence: the buffer we are about to overwrite was consumed via
            // ds_loads last iteration; async LDS writes are unordered vs those
            // reads, so make sure all prior DS reads have drained first.
            asm volatile("s_wait_dscnt 0x0" ::: "memory");
            ASYNC_PREFETCH(c + 1);
            // up to 8 outstanding; <=4 means chunk c (issued first) is done
            asm volatile("s_wait_asynccnt 0x4" ::: "memory");
        } else {
            asm volatile("s_wait_asynccnt 0x0" ::: "memory");
        }

        const int p   = c & 1;
        const int buf = c & 1;
        const float4* fp = (const float4*)&lbuf[w][buf][0][0];
        const float4 mn  = fp[lane];
        const float4 lsv = fp[32 + lane];
        const float4 nev = fp[64 + lane];
        const float4 npv = fp[96 + lane];

        // enc = mean + exp(0.5*log_std)*noise  (stop_gradient irrelevant fwd)
        float e0 = mn.x + __expf(0.5f * lsv.x) * nev.x;
        float e1 = mn.y + __expf(0.5f * lsv.y) * nev.y;
        float e2 = mn.z + __expf(0.5f * lsv.z) * nev.z;
        float e3 = mn.w + __expf(0.5f * lsv.w) * nev.w;

        // prior = fixed mean + noise; mean_1 = e_13, mean_2 = 1 - e_13
        const int j0 = lane * 4;
        const float base = p ? 1.0f : 0.0f;
        const float flip = 1.0f - base;
        float q0 = ((j0 + 0) == 13 ? flip : base) + npv.x;
        float q1 = ((j0 + 1) == 13 ? flip : base) + npv.y;
        float q2 = ((j0 + 2) == 13 ? flip : base) + npv.z;
        float q3 = ((j0 + 3) == 13 ? flip : base) + npv.w;

        // row max (enc, prior) across 32 lanes
        float me = fmaxf(fmaxf(e0, e1), fmaxf(e2, e3));
        float mp = fmaxf(fmaxf(q0, q1), fmaxf(q2, q3));
#pragma unroll
        for (int o = 16; o > 0; o >>= 1) {
            me = fmaxf(me, __shfl_xor(me, o, 32));
            mp = fmaxf(mp, __shfl_xor(mp, o, 32));
        }

        float t0 = __expf(e0 - me), t1 = __expf(e1 - me);
        float t2 = __expf(e2 - me), t3 = __expf(e3 - me);
        float u0 = __expf(q0 - mp), u1 = __expf(q1 - mp);
        float u2 = __expf(q2 - mp), u3 = __expf(q3 - mp);
        float se = (t0 + t1) + (t2 + t3);
        float sp = (u0 + u1) + (u2 + u3);
#pragma unroll
        for (int o = 16; o > 0; o >>= 1) {
            se += __shfl_xor(se, o, 32);
            sp += __shfl_xor(sp, o, 32);
        }

        const float ce  = me + __logf(se);   // logsumexp(enc)
        const float cp  = mp + __logf(sp);   // logsumexp(prior)
        const float inv = 1.0f / se;

        // sum_j q_j * (log q_j - log p_j), q = softmax(enc), logp = logsoftmax(prior)
        float acc = t0 * inv * ((e0 - ce) - (q0 - cp))
                  + t1 * inv * ((e1 - ce) - (q1 - cp))
                  + t2 * inv * ((e2 - ce) - (q2 - cp))
                  + t3 * inv * ((e3 - ce) - (q3 - cp));
#pragma unroll
        for (int o = 16; o > 0; o >>= 1) acc += __shfl_xor(acc, o, 32);

        if (p == 0) zAcc0 += acc; else zAcc1 += acc;
    }
#undef ASYNC_PREFETCH

    if (lane == 0) { wz[w][0] = zAcc0; wz[w][1] = zAcc1; }
    __syncthreads();
    if (tid == 0) {
        float st = 0.0f, ss = 0.0f;
#pragma unroll
        for (int i = 0; i < WAVES_PER_BLOCK; ++i) { st += wz[i][0]; ss += wz[i][1]; }
        zt_part[blockIdx.x] = st;
        zs_part[blockIdx.x] = ss;
    }
}

// ---------------------------------------------------------------------------
// Kernel 2: BCE-with-logits term and softmax-entropy term (S=2), block partials.
// ---------------------------------------------------------------------------
__global__ __launch_bounds__(256) void small_terms_kernel(
    const float* __restrict__ y_zt, const float* __restrict__ target,
    const float* __restrict__ s_zt,
    float* __restrict__ lt_part, float* __restrict__ le_part)
{
    __shared__ float s1[256], s2[256];
    const int tid = threadIdx.x;
    const int i = blockIdx.x * 256 + tid;

    const float y = y_zt[i];
    const float t = target[i];
    const float lt = fmaxf(y, 0.0f) - y * t + log1pf(__expf(-fabsf(y)));

    const float a = s_zt[2 * i + 0];
    const float b = s_zt[2 * i + 1];
    const float m  = fmaxf(a, b);
    const float ea = __expf(a - m), eb = __expf(b - m);
    const float s  = ea + eb;
    const float ls = __logf(s);
    const float la = a - m - ls, lb = b - m - ls;
    const float le = -((ea / s) * la + (eb / s) * lb);

    s1[tid] = lt; s2[tid] = le;
    __syncthreads();
#pragma unroll
    for (int o = 128; o > 0; o >>= 1) {
        if (tid < o) { s1[tid] += s1[tid + o]; s2[tid] += s2[tid + o]; }
        __syncthreads();
    }
    if (tid == 0) { lt_part[blockIdx.x] = s1[0]; le_part[blockIdx.x] = s2[0]; }
}

// ---------------------------------------------------------------------------
// Kernel 3: deterministic final reduction + lambda schedule + scalar output.
// ---------------------------------------------------------------------------
__global__ __launch_bounds__(256) void final_kernel(
    const float* __restrict__ zt_part, const float* __restrict__ zs_part,
    const float* __restrict__ lt_part, const float* __restrict__ le_part,
    const int* __restrict__ step, float* __restrict__ out)
{
    __shared__ float sa[256], sb[256], sc[256];
    const int tid = threadIdx.x;
    float z = 0.0f, lt = 0.0f, le = 0.0f;
    for (int i = tid; i < NB1; i += 256) z += zt_part[i] + zs_part[i];
    for (int i = tid; i < NB2; i += 256) { lt += lt_part[i]; le += le_part[i]; }
    sa[tid] = z; sb[tid] = lt; sc[tid] = le;
    __syncthreads();
#pragma unroll
    for (int o = 128; o > 0; o >>= 1) {
        if (tid < o) {
            sa[tid] += sa[tid + o];
            sb[tid] += sb[tid + o];
            sc[tid] += sc[tid + o];
        }
        __syncthreads();
    }
    if (tid == 0) {
        const float invB   = 1.0f / (float)B_ROWS;
        const float frac   = (float)step[0] / 30.0f;
        const float lam_e  = 0.1f   * exp2f(frac);         // 2.0 ** frac
        const float lam_od = 0.036f * powf(1.1f, frac);    // 1.1 ** frac
        out[0] = sb[0] * invB + lam_od * (sa[0] * invB) + lam_e * (sc[0] * invB);
    }
}

// ---------------------------------------------------------------------------
// Launch: inputs in setup_inputs() order:
// 0 mean_t, 1 mean_s, 2 log_std_t, 3 log_std_s, 4 y_zt, 5 s_zt, 6 s_zs(unused),
// 7 z1(unused), 8 z2(unused), 9 target, 10 noise_pt, 11 noise_ps,
// 12 noise_et, 13 noise_es, 14 current_step(int scalar)
// ---------------------------------------------------------------------------
extern "C" void kernel_launch(void* const* d_in, const int* in_sizes, int n_in,
                              void* d_out, int out_size, void* d_ws, size_t ws_size,
                              hipStream_t stream) {
    const float* mean_t    = (const float*)d_in[0];
    const float* mean_s    = (const float*)d_in[1];
    const float* log_std_t = (const float*)d_in[2];
    const float* log_std_s = (const float*)d_in[3];
    const float* y_zt      = (const float*)d_in[4];
    const float* s_zt      = (const float*)d_in[5];
    const float* target    = (const float*)d_in[9];
    const float* noise_pt  = (const float*)d_in[10];
    const float* noise_ps  = (const float*)d_in[11];
    const float* noise_et  = (const float*)d_in[12];
    const float* noise_es  = (const float*)d_in[13];
    const int*   step      = (const int*)d_in[14];

    float* ws      = (float*)d_ws;
    float* zt_part = ws;                  // NB1 floats
    float* zs_part = ws + NB1;            // NB1 floats
    float* lt_part = ws + 2 * NB1;        // NB2 floats
    float* le_part = ws + 2 * NB1 + NB2;  // NB2 floats

    kl_rows_kernel<<<NB1, 256, 0, stream>>>(
        mean_t, mean_s, log_std_t, log_std_s,
        noise_pt, noise_ps, noise_et, noise_es, zt_part, zs_part);
    small_terms_kernel<<<NB2, 256, 0, stream>>>(y_zt, target, s_zt, lt_part, le_part);
    final_kernel<<<1, 256, 0, stream>>>(zt_part, zs_part, lt_part, le_part, step,
                                        (float*)d_out);
}